// GCN_Model_3822520893927
// MI455X (gfx1250) — compile-verified
//
#include <hip/hip_runtime.h>
#include <hip/hip_bf16.h>

// GCN 3-layer: per layer  Y = scatter_add_{dst}( (H @ W)[src] ) + b
// N=100000 nodes, E=1600000 edges, channels 128 -> 64 -> 64 -> 64.
// GEMMs: V_WMMA_F32_16X16X4_F32 (exact fp32 WMMA path on CDNA5).
// Scatter: b128 row gathers + hardware global_atomic_add_f32; both the
// gathered xw (25.6MB) and the destination h (25.6MB) are L2-resident
// (192MB L2), so the scatter runs at L2/atomic speed, not HBM speed.

#define N_NODES 100000
#define N_EDGES 1600000
#define OUTC    64

typedef float v2f __attribute__((ext_vector_type(2)));
typedef float v8f __attribute__((ext_vector_type(8)));

// ---------------------------------------------------------------------------
// out[n*64 + c] = b[c]  -- bias broadcast via b128 stores.
// total4 = N_NODES*64/4 float4 elements.
// ---------------------------------------------------------------------------
__global__ void gcn_bias_init(float4* __restrict__ out,
                              const float4* __restrict__ b4,
                              int total4) {
    int idx = blockIdx.x * blockDim.x + threadIdx.x;
    if (idx < total4) {
        out[idx] = b4[idx & (OUTC / 4 - 1)];
    }
}

// ---------------------------------------------------------------------------
// Y[M x 64] = X[M x K] @ W[K x 64], fp32 WMMA 16x16x4.
// Block = 256 threads = 8 waves. Wave w: n-tile = w&3, m-subtile = w>>2.
// Block covers 32 rows x 64 cols. Grid.x = M/32 (M = 100000 = 3125*32).
// K must be a multiple of 4 (128 or 64 here).
// Fragment layouts (lane L, m=L&15, h=L>>4):
//   A vgpr v : X[m][k0 + 2h + v]        B vgpr v : W[k0 + 2h + v][n]
//   D vgpr r : row (mbase + r + 8h), col n
// ---------------------------------------------------------------------------
__global__ void gcn_gemm_wmma(const float* __restrict__ X,
                              const float* __restrict__ W,
                              float* __restrict__ Y,
                              int K) {
    const int lane  = threadIdx.x & 31;
    const int wave  = threadIdx.x >> 5;
    const int ntile = wave & 3;                       // 4 n-tiles of 16
    const int mbase = (blockIdx.x * 2 + (wave >> 2)) * 16;

    const int m = lane & 15;
    const int h = lane >> 4;
    const int n = ntile * 16 + (lane & 15);

    const float* __restrict__ xrow = X + (long long)(mbase + m) * K;

    v8f acc = {0.f, 0.f, 0.f, 0.f, 0.f, 0.f, 0.f, 0.f};

    for (int k0 = 0; k0 < K; k0 += 4) {
        const int ka = k0 + 2 * h;                    // even -> 8B aligned
        v2f a = *(const v2f*)(xrow + ka);
        v2f b;
        b.x = W[(long long)ka * OUTC + n];
        b.y = W[(long long)(ka + 1) * OUTC + n];
        // 8 args: (neg_a, A, neg_b, B, c_mod, C, reuse_a, reuse_b)
        acc = __builtin_amdgcn_wmma_f32_16x16x4_f32(
            false, a, false, b, (short)0, acc, false, false);
    }

#pragma unroll
    for (int r = 0; r < 8; ++r) {
        Y[(long long)(mbase + r + 8 * h) * OUTC + n] = acc[r];
    }
}

// ---------------------------------------------------------------------------
// For each edge e:  out[dst[e]*64 + c] += xw[src[e]*64 + c], c = 0..63.
// One thread per (edge, 4-channel quad): e = idx>>4, q = idx&15.
// 16 consecutive threads cover one edge's 256B row -> coalesced b128 gather,
// then 4 hardware f32 atomics. Index loads amortized 4x vs per-channel.
// ---------------------------------------------------------------------------
__global__ void gcn_scatter_add(const float* __restrict__ xw,
                                const int* __restrict__ src,
                                const int* __restrict__ dst,
                                float* __restrict__ out) {
    long long idx = (long long)blockIdx.x * blockDim.x + threadIdx.x;
    long long e = idx >> 4;
    int       q = (int)(idx & 15);                    // quad index: 4 floats
    if (e < N_EDGES) {
        int s = src[e];
        int d = dst[e];
        float4 v = *(const float4*)(xw + (long long)s * OUTC + q * 4);
        float* o = out + (long long)d * OUTC + q * 4;
        unsafeAtomicAdd(o + 0, v.x);
        unsafeAtomicAdd(o + 1, v.y);
        unsafeAtomicAdd(o + 2, v.z);
        unsafeAtomicAdd(o + 3, v.w);
    }
}

// ---------------------------------------------------------------------------
extern "C" void kernel_launch(void* const* d_in, const int* in_sizes, int n_in,
                              void* d_out, int out_size, void* d_ws, size_t ws_size,
                              hipStream_t stream) {
    (void)in_sizes; (void)n_in; (void)out_size; (void)ws_size;

    const float* x   = (const float*)d_in[0];
    const int*   ei  = (const int*)d_in[1];     // edge_index [2, E] row-major
    const float* W1  = (const float*)d_in[2];
    const float* b1  = (const float*)d_in[3];
    const float* W2  = (const float*)d_in[4];
    const float* b2  = (const float*)d_in[5];
    const float* W3  = (const float*)d_in[6];
    const float* b3  = (const float*)d_in[7];

    const int* src = ei;
    const int* dst = ei + N_EDGES;

    float* h1 = (float*)d_out;
    float* h2 = h1 + (long long)N_NODES * OUTC;
    float* h3 = h2 + (long long)N_NODES * OUTC;

    float* xw = (float*)d_ws;                   // [N_NODES, 64] scratch, 25.6MB

    const int featTotal4  = N_NODES * OUTC / 4; // 1.6M float4s
    const int initBlocks  = (featTotal4 + 255) / 256;
    const int gemmBlocks  = N_NODES / 32;       // 3125, exact
    const long long scatThreads = (long long)N_EDGES * 16;  // 25.6M
    const int scatBlocks  = (int)((scatThreads + 255) / 256);

    // ---- Layer 1: x[100000x128] @ W1[128x64] ----
    gcn_bias_init <<<initBlocks, 256, 0, stream>>>((float4*)h1, (const float4*)b1, featTotal4);
    gcn_gemm_wmma <<<gemmBlocks, 256, 0, stream>>>(x, W1, xw, 128);
    gcn_scatter_add<<<scatBlocks, 256, 0, stream>>>(xw, src, dst, h1);

    // ---- Layer 2: h1[100000x64] @ W2[64x64] ----
    gcn_bias_init <<<initBlocks, 256, 0, stream>>>((float4*)h2, (const float4*)b2, featTotal4);
    gcn_gemm_wmma <<<gemmBlocks, 256, 0, stream>>>(h1, W2, xw, 64);
    gcn_scatter_add<<<scatBlocks, 256, 0, stream>>>(xw, src, dst, h2);

    // ---- Layer 3: h2[100000x64] @ W3[64x64] ----
    gcn_bias_init <<<initBlocks, 256, 0, stream>>>((float4*)h3, (const float4*)b3, featTotal4);
    gcn_gemm_wmma <<<gemmBlocks, 256, 0, stream>>>(h2, W3, xw, 64);
    gcn_scatter_add<<<scatBlocks, 256, 0, stream>>>(xw, src, dst, h3);
}